// BlockConvolution_1_31911607009797
// MI455X (gfx1250) — compile-verified
//
#include <hip/hip_runtime.h>

// ---------------- problem constants (from reference) ----------------
#define BATCH   8192
#define NN      40
#define FDIM    256
#define NFEAT   (NN * FDIM)          // 10240 batchnorm features
#define ROWS    (BATCH * NN)         // 327680 flattened rows
#define WG_ROWS 128                  // rows per workgroup in GEMM kernel
#define TSTR    260                  // padded LDS stride for T tile (floats)
#define KCHUNK  64                   // K rows of W staged per iteration
#define WSTR    544                  // dwords per swizzled k-pair row of W

typedef float v2f __attribute__((ext_vector_type(2)));
typedef float v8f __attribute__((ext_vector_type(8)));

// =====================================================================
// Kernel 1: fused  Y = (blockdiag(adj) @ input) @ W
//   Phase 1 (VALU): T rows into LDS.  10-row "blocks" of the flattened
//   row index are exactly the adj diagonal blocks, and they read exactly
//   their own 10 input rows -> perfect locality.
//   Phase 2 (WMMA): T[128x256] @ W[256x256] with v_wmma_f32_16x16x4_f32.
// =====================================================================
extern "C" __global__ __launch_bounds__(256)
void k1_fused_gemm(const float* __restrict__ input,
                   const float* __restrict__ adj,
                   const float* __restrict__ W,
                   float* __restrict__ Y)
{
    extern __shared__ float smem[];
    float* Tsh = smem;                       // 128 x 260 floats
    float* Wsh = smem + WG_ROWS * TSTR;      // 32 pair-rows x 544 dwords

    const int t    = threadIdx.x;
    const int lane = t & 31;
    const int wv   = t >> 5;                 // wave id 0..7, owns 16-row strip
    const int half = lane >> 4;              // lane group (ISA fragment layout)
    const int l16  = lane & 15;
    const int R0   = blockIdx.x * WG_ROWS;

    // ---------------- Phase 1: T = blockdiag(adj) * input ----------------
    {
        const int gb0 = R0 / 10;
        const int gb1 = (R0 + WG_ROWS - 1) / 10;
        for (int gb = gb0; gb <= gb1; ++gb) {
            // stage this block's 10 input values for column t in registers
            float in[10];
            const float* ip = input + (size_t)gb * 10 * FDIM + t;
            #pragma unroll
            for (int j = 0; j < 10; ++j) in[j] = ip[j * FDIM];

            int r0 = gb * 10;       if (r0 < R0) r0 = R0;
            int r1 = gb * 10 + 10;  if (r1 > R0 + WG_ROWS) r1 = R0 + WG_ROWS;
            const int abase = (gb & 3) * 10;   // block index within batch * 10
            for (int r = r0; r < r1; ++r) {
                const float* arow = adj + (size_t)r * NN + abase; // uniform -> s_load
                float acc = 0.f;
                #pragma unroll
                for (int j = 0; j < 10; ++j) acc = fmaf(arow[j], in[j], acc);
                Tsh[(r - R0) * TSTR + t] = acc;
            }
        }
    }
    __syncthreads();

    // ---------------- Phase 2: Y = T @ W via WMMA ----------------
    v8f acc[16];
    #pragma unroll
    for (int c = 0; c < 16; ++c)
        #pragma unroll
        for (int j = 0; j < 8; ++j) acc[c][j] = 0.f;

    const int arow = wv * 16 + l16;          // A-fragment row in T

    for (int kc = 0; kc < FDIM; kc += KCHUNK) {
        // cooperative stage of W[kc..kc+63][:], pair-swizzled so a B-fragment
        // (two consecutive K values, same column) is one aligned b64 LDS read.
        for (int it = 0; it < KCHUNK; ++it) {            // row = it, col = t
            float v = W[(size_t)(kc + it) * FDIM + t];   // coalesced
            Wsh[(it >> 1) * WSTR + t * 2 + (it & 1)] = v;
        }
        __syncthreads();

        for (int kk = 0; kk < KCHUNK; kk += 4) {
            // A fragment: lane<16 holds K=k0,k0+1 ; lane>=16 holds K=k0+2,k0+3
            v2f a = *(const v2f*)(Tsh + arow * TSTR + (kc + kk) + half * 2);
            const float* wbase = Wsh + ((kk >> 1) + half) * WSTR + l16 * 2;
            #pragma unroll
            for (int c = 0; c < 16; ++c) {
                v2f b = *(const v2f*)(wbase + c * 32);
                acc[c] = __builtin_amdgcn_wmma_f32_16x16x4_f32(
                    false, a, false, b, (short)0, acc[c], false, false);
            }
        }
        __syncthreads();
    }

    // ---------------- store Y (C/D layout: VGPR j -> M=j / j+8) ----------------
    #pragma unroll
    for (int c = 0; c < 16; ++c) {
        #pragma unroll
        for (int j = 0; j < 8; ++j) {
            int m = half * 8 + j;
            Y[(size_t)(R0 + wv * 16 + m) * FDIM + c * 16 + l16] = acc[c][j];
        }
    }
}

// =====================================================================
// Kernel 2: per-feature sum / sum-of-squares over the batch axis
// =====================================================================
extern "C" __global__ __launch_bounds__(256)
void k2_reduce(const float* __restrict__ Y,
               float* __restrict__ sum, float* __restrict__ sumsq)
{
    const int n     = blockIdx.x % NN;
    const int chunk = blockIdx.x / NN;       // 64 chunks of 128 batches
    const int t     = threadIdx.x;           // feature column f
    float s = 0.f, q = 0.f;
    const int b0 = chunk * 128;
    for (int b = b0; b < b0 + 128; ++b) {
        float v = Y[(size_t)(b * NN + n) * FDIM + t];  // 1KB coalesced rows
        s += v;
        q = fmaf(v, v, q);
    }
    atomicAdd(&sum[n * FDIM + t],   s);
    atomicAdd(&sumsq[n * FDIM + t], q);
}

// =====================================================================
// Kernel 3: fold mean/var/gamma/beta into per-feature scale & shift
//   out = (Y-mean)*rsqrt(var+eps)*gamma[blk] + sum_i beta[i]
// =====================================================================
extern "C" __global__
void k2b_finalize(const float* __restrict__ sum, const float* __restrict__ sumsq,
                  const float* __restrict__ gamma, const float* __restrict__ beta,
                  float* __restrict__ scale, float* __restrict__ shift)
{
    const int feat = blockIdx.x * 256 + threadIdx.x;   // 0..10239
    const int n    = feat >> 8;
    const int blk  = n / 10;
    const float inv  = 1.0f / (float)BATCH;
    const float mean = sum[feat] * inv;
    float var        = fmaxf(sumsq[feat] * inv - mean * mean, 0.f);
    const float rs   = rsqrtf(var + 1e-5f);
    const float sc   = gamma[blk * NFEAT + feat] * rs;
    const float sh   = -mean * sc
                     + beta[0 * NFEAT + feat] + beta[1 * NFEAT + feat]
                     + beta[2 * NFEAT + feat] + beta[3 * NFEAT + feat];
    scale[feat] = sc;
    shift[feat] = sh;
}

// =====================================================================
// Kernel 4: in-place normalize, float4 vectorized
// =====================================================================
extern "C" __global__ __launch_bounds__(256)
void k3_norm(float* __restrict__ out,
             const float* __restrict__ scale, const float* __restrict__ shift)
{
    const size_t base = ((size_t)blockIdx.x * 256 + threadIdx.x) * 4;
    const int feat = (int)(base % NFEAT);    // NFEAT % 4 == 0 -> aligned
    float4 y        = *(const float4*)(out + base);
    const float4 sc = *(const float4*)(scale + feat);
    const float4 sh = *(const float4*)(shift + feat);
    float4 r;
    r.x = fmaf(y.x, sc.x, sh.x);
    r.y = fmaf(y.y, sc.y, sh.y);
    r.z = fmaf(y.z, sc.z, sh.z);
    r.w = fmaf(y.w, sc.w, sh.w);
    *(float4*)(out + base) = r;
}

// =====================================================================
extern "C" void kernel_launch(void* const* d_in, const int* in_sizes, int n_in,
                              void* d_out, int out_size, void* d_ws, size_t ws_size,
                              hipStream_t stream)
{
    const float* input = (const float*)d_in[0];
    const float* adj   = (const float*)d_in[1];
    const float* W     = (const float*)d_in[2];
    const float* gamma = (const float*)d_in[3];
    const float* beta  = (const float*)d_in[4];
    float* out = (float*)d_out;

    float* ws    = (float*)d_ws;
    float* sum   = ws;
    float* sumsq = ws + NFEAT;
    float* scale = ws + 2 * NFEAT;
    float* shift = ws + 3 * NFEAT;           // total 160KB scratch

    hipMemsetAsync(sum, 0, 2 * NFEAT * sizeof(float), stream);

    const size_t shmem = (size_t)(WG_ROWS * TSTR + (KCHUNK / 2) * WSTR) * sizeof(float);
    k1_fused_gemm<<<ROWS / WG_ROWS, 256, shmem, stream>>>(input, adj, W, out);
    k2_reduce<<<NN * (BATCH / 128), 256, 0, stream>>>(out, sum, sumsq);
    k2b_finalize<<<NFEAT / 256, 256, 0, stream>>>(sum, sumsq, gamma, beta, scale, shift);
    k3_norm<<<(ROWS * (FDIM / 4)) / 256, 256, 0, stream>>>(out, scale, shift);
}